// GAT_dialoggcn_44727789421070
// MI455X (gfx1250) — compile-verified
//
#include <hip/hip_runtime.h>
#include <hip/hip_bf16.h>

// Problem: B=16, N=8192, D=128
#define BB 16
#define NN 8192
#define DD 128
#define PP 32                  // blocks per batch for streaming passes
#define ROWS (NN / PP)         // 256 rows per block
#define MASK_NEG 1e30f

typedef float v2f __attribute__((ext_vector_type(2)));
typedef float v4f __attribute__((ext_vector_type(4)));
typedef float v8f __attribute__((ext_vector_type(8)));

__device__ __forceinline__ float wave_sum32(float v) {
#pragma unroll
    for (int m = 16; m > 0; m >>= 1) v += __shfl_xor(v, m, 32);
    return v;
}

__device__ __forceinline__ v4f ldg_nt(const float* p) {
    // single-use stream: non-temporal 128-bit load
    return __builtin_nontemporal_load((const v4f*)p);
}

// ---------------------------------------------------------------------------
// Pass 1: alpha[b,n] = Q[b].wq + K[b,n].wk + bias - (1-adj)*1e30
// Grid: BB*PP blocks x 256 threads (8 wave32). Wave-per-row, lane l covers
// d = 4l..4l+3 (coalesced 512B per wave per row). Writes alpha into the
// attn_weight slot of d_out (scratch; overwritten by pass 3) and a per-block
// max into the workspace.
// ---------------------------------------------------------------------------
__global__ __launch_bounds__(256)
void k_alpha(const float* __restrict__ Q, const float* __restrict__ K,
             const float* __restrict__ adj, const float* __restrict__ lin_w,
             const float* __restrict__ lin_b,
             float* __restrict__ alpha_out, float* __restrict__ blockmax) {
    const int b    = blockIdx.x >> 5;        // / PP
    const int blk  = blockIdx.x & (PP - 1);  // % PP
    const int tid  = threadIdx.x;
    const int lane = tid & 31;
    const int wave = tid >> 5;

    __shared__ float sh_red[8];

    const v4f wq4 = *(const v4f*)(lin_w + 4 * lane);
    const v4f wk4 = *(const v4f*)(lin_w + DD + 4 * lane);
    const v4f q4  = *(const v4f*)(Q + (size_t)b * DD + 4 * lane);

    float qp = q4.x * wq4.x + q4.y * wq4.y + q4.z * wq4.z + q4.w * wq4.w;
    const float qdot = wave_sum32(qp) + lin_b[0];

    const int n_base = blk * ROWS;
    const float* Kb  = K + ((size_t)b * NN + n_base) * DD;

    float wmax = -INFINITY;
    for (int r = wave; r < ROWS; r += 8) {
        const v4f k4 = ldg_nt(Kb + (size_t)r * DD + 4 * lane);
        float p = k4.x * wk4.x + k4.y * wk4.y + k4.z * wk4.z + k4.w * wk4.w;
        p = wave_sum32(p);                   // all lanes hold the full dot
        const int n   = n_base + r;
        const float a = qdot + p - (1.0f - adj[(size_t)b * NN + n]) * MASK_NEG;
        if (lane == 0) alpha_out[(size_t)b * NN + n] = a;
        wmax = fmaxf(wmax, a);
    }
    if (lane == 0) sh_red[wave] = wmax;
    __syncthreads();
    if (tid == 0) {
        float m = sh_red[0];
#pragma unroll
        for (int i = 1; i < 8; ++i) m = fmaxf(m, sh_red[i]);
        blockmax[blockIdx.x] = m;            // [b*PP + blk]
    }
}

// ---------------------------------------------------------------------------
// Pass 2: per batch, global max over PP block maxes + sum of exp.
// Grid: BB blocks x 256 threads. Reads alpha (32 KB/batch) from d_out.
// rowstats[b] = {gmax, 1/sum}.
// ---------------------------------------------------------------------------
__global__ __launch_bounds__(256)
void k_stats(const float* __restrict__ alpha, const float* __restrict__ blockmax,
             float* __restrict__ rowstats) {
    const int b    = blockIdx.x;
    const int tid  = threadIdx.x;
    const int lane = tid & 31;
    const int wave = tid >> 5;

    __shared__ float sh_red[8];
    __shared__ float sh_max;

    if (tid == 0) {
        float m = blockmax[b * PP];
#pragma unroll
        for (int i = 1; i < PP; ++i) m = fmaxf(m, blockmax[b * PP + i]);
        sh_max = m;
    }
    __syncthreads();
    const float gmax = sh_max;

    float lsum = 0.0f;
    for (int n = tid; n < NN; n += 256)
        lsum += __expf(alpha[(size_t)b * NN + n] - gmax);
    lsum = wave_sum32(lsum);
    if (lane == 0) sh_red[wave] = lsum;
    __syncthreads();
    if (tid == 0) {
        float s = 0.0f;
#pragma unroll
        for (int i = 0; i < 8; ++i) s += sh_red[i];
        rowstats[2 * b + 0] = gmax;
        rowstats[2 * b + 1] = 1.0f / s;
    }
}

// ---------------------------------------------------------------------------
// Pass 3: w = exp(alpha-gmax)*inv  (written over alpha in d_out), and the
// per-block partial weighted V sums:
//   partS[b][blk][r*128+d] = sum_{n in chunk} w_n * onehot[n,r] * V[b,n,d]
// Deterministic: per-wave register accumulators staged to LDS, summed over
// the 8 waves in fixed order.
// ---------------------------------------------------------------------------
__global__ __launch_bounds__(256)
void k_norm_s(const float* __restrict__ V, const float* __restrict__ onehot,
              const float* __restrict__ rowstats, float* __restrict__ attn_w,
              float* __restrict__ partS) {
    const int b    = blockIdx.x >> 5;
    const int blk  = blockIdx.x & (PP - 1);
    const int tid  = threadIdx.x;
    const int lane = tid & 31;
    const int wave = tid >> 5;

    __shared__ float sh_part[8][2 * DD];     // 8 KB

    const float gmax = rowstats[2 * b + 0];
    const float inv  = rowstats[2 * b + 1];

    const int n_base = blk * ROWS;
    const float* Vb  = V + ((size_t)b * NN + n_base) * DD;
    const float* Mb  = onehot + ((size_t)b * NN + n_base) * 2;

    v4f acc0 = {0.f, 0.f, 0.f, 0.f};
    v4f acc1 = {0.f, 0.f, 0.f, 0.f};
    for (int r = wave; r < ROWS; r += 8) {
        const int n   = n_base + r;
        const float a = attn_w[(size_t)b * NN + n];        // alpha scratch
        const float w = __expf(a - gmax) * inv;
        if (lane == 0) attn_w[(size_t)b * NN + n] = w;     // final attn_weight
        const float c0 = w * Mb[2 * r + 0];
        const float c1 = w * Mb[2 * r + 1];
        const v4f v4 = ldg_nt(Vb + (size_t)r * DD + 4 * lane);
        acc0.x = fmaf(c0, v4.x, acc0.x); acc0.y = fmaf(c0, v4.y, acc0.y);
        acc0.z = fmaf(c0, v4.z, acc0.z); acc0.w = fmaf(c0, v4.w, acc0.w);
        acc1.x = fmaf(c1, v4.x, acc1.x); acc1.y = fmaf(c1, v4.y, acc1.y);
        acc1.z = fmaf(c1, v4.z, acc1.z); acc1.w = fmaf(c1, v4.w, acc1.w);
    }
    float* wp = &sh_part[wave][0];
    wp[0 * DD + 4 * lane + 0] = acc0.x; wp[0 * DD + 4 * lane + 1] = acc0.y;
    wp[0 * DD + 4 * lane + 2] = acc0.z; wp[0 * DD + 4 * lane + 3] = acc0.w;
    wp[1 * DD + 4 * lane + 0] = acc1.x; wp[1 * DD + 4 * lane + 1] = acc1.y;
    wp[1 * DD + 4 * lane + 2] = acc1.z; wp[1 * DD + 4 * lane + 3] = acc1.w;
    __syncthreads();

    float s = 0.0f;
#pragma unroll
    for (int wv = 0; wv < 8; ++wv) s += sh_part[wv][tid];
    partS[((size_t)b * PP + blk) * (2 * DD) + tid] = s;
}

// ---------------------------------------------------------------------------
// Pass 4: reduce partS (fixed order) into A = S(16x256) in LDS, then
// attn_sum(16x128) = A @ Rel(256x128) with V_WMMA_F32_16X16X4_F32.
// Rel[k][e] = rel_emb[r][d][e], k = r*128+d -> flat k*128+e.
// One block, 8 waves; wave w -> output columns [16w,16w+16).
// Lane layout (ISA 7.12.2, f32):
//   A 16x4 : lanes 0-15: M=lane, K=k+0/k+1; lanes 16-31: M=lane-16, K=k+2/k+3
//   B 4x16 : lanes 0-15: N=lane, K=k+0/k+1; lanes 16-31: N=lane-16, K=k+2/k+3
//   C/D    : VGPR i -> M = i + 8*(lane>=16), N = lane&15
// ---------------------------------------------------------------------------
__global__ __launch_bounds__(256)
void gat_phase2(const float* __restrict__ partS, const float* __restrict__ rel_emb,
                float* __restrict__ attn_sum) {
    const int tid   = threadIdx.x;
    const int lane  = tid & 31;
    const int wave  = tid >> 5;
    const int n0    = wave * 16;
    const int m     = lane & 15;
    const int khalf = (lane >> 4) * 2;

    __shared__ float shS[BB * 2 * DD];       // 16 KB: A-matrix rows b, cols k

    for (int idx = tid; idx < BB * 2 * DD; idx += 256) {
        const int bb = idx >> 8;             // batch (row)
        const int k  = idx & 255;
        float s = 0.0f;
#pragma unroll 8
        for (int p = 0; p < PP; ++p)
            s += partS[((size_t)bb * PP + p) * (2 * DD) + k];
        shS[idx] = s;
    }
    __syncthreads();

    v8f acc = {};
#pragma unroll 4
    for (int k = 0; k < 2 * DD; k += 4) {
        v2f a, bm;
        a.x  = shS[m * (2 * DD) + k + khalf + 0];
        a.y  = shS[m * (2 * DD) + k + khalf + 1];
        bm.x = rel_emb[(size_t)(k + khalf + 0) * DD + n0 + m];
        bm.y = rel_emb[(size_t)(k + khalf + 1) * DD + n0 + m];
        acc = __builtin_amdgcn_wmma_f32_16x16x4_f32(
            /*neg_a=*/false, a, /*neg_b=*/false, bm,
            /*c_mod=*/(short)0, acc, /*reuse_a=*/false, /*reuse_b=*/false);
    }

    const int rowAdd = (lane >> 4) * 8;
#pragma unroll
    for (int i = 0; i < 8; ++i)
        attn_sum[(size_t)(i + rowAdd) * DD + n0 + m] = acc[i];
}

extern "C" void kernel_launch(void* const* d_in, const int* in_sizes, int n_in,
                              void* d_out, int out_size, void* d_ws, size_t ws_size,
                              hipStream_t stream) {
    const float* Q       = (const float*)d_in[0];
    const float* K       = (const float*)d_in[1];
    const float* V       = (const float*)d_in[2];
    const float* adj     = (const float*)d_in[3];
    const float* onehot  = (const float*)d_in[4];
    const float* lin_w   = (const float*)d_in[5];
    const float* lin_b   = (const float*)d_in[6];
    const float* rel_emb = (const float*)d_in[7];

    float* attn_w   = (float*)d_out;                     // B*N (alpha scratch -> weights)
    float* attn_sum = (float*)d_out + (size_t)BB * NN;   // B*D

    float* partS    = (float*)d_ws;                      // BB*PP*2*DD floats (512 KB)
    float* blockmax = partS + (size_t)BB * PP * 2 * DD;  // BB*PP floats
    float* rowstats = blockmax + (size_t)BB * PP;        // 2*BB floats

    k_alpha  <<<BB * PP, 256, 0, stream>>>(Q, K, adj, lin_w, lin_b, attn_w, blockmax);
    k_stats  <<<BB,      256, 0, stream>>>(attn_w, blockmax, rowstats);
    k_norm_s <<<BB * PP, 256, 0, stream>>>(V, onehot, rowstats, attn_w, partS);
    gat_phase2<<<1,      256, 0, stream>>>(partS, rel_emb, attn_sum);
}